// EpipolarAttention_9302899163604
// MI455X (gfx1250) — compile-verified
//
#include <hip/hip_runtime.h>
#include <math.h>

// ---------------------------------------------------------------------------
// EpipolarAttention, fused flash-style implementation for gfx1250 (MI455X).
//   B=2, C=3, H=W=64, N=4096.  Never materializes any (N x N) intermediate.
//   A[i,k] tiles are produced by V_WMMA_F32_16X16X4_F32 (C=3 padded to K=4).
//   Fast-math intrinsics (v_sqrt_f32 / v_rcp_f32 / v_rsq_f32) keep the
//   N^2-swept inner loops tight: the softmax with sharpness 50 makes
//   last-ulp sqrt precision irrelevant.
// ---------------------------------------------------------------------------

#define NPIX 4096
#define BATCH 2

typedef __attribute__((ext_vector_type(2))) float v2f;
typedef __attribute__((ext_vector_type(8))) float v8f;

__device__ __forceinline__ float fast_sqrt(float x) { return __builtin_amdgcn_sqrtf(x); }
__device__ __forceinline__ float fast_rcp(float x)  { return __builtin_amdgcn_rcpf(x); }
__device__ __forceinline__ float fast_rsq(float x)  { return __builtin_amdgcn_rsqf(x); }

__device__ __forceinline__ void softmax_combine(float& m, float& l, float om, float ol) {
    float nm = fmaxf(m, om);
    l = l * __expf(m - nm) + ol * __expf(om - nm);
    m = nm;
}

__device__ __forceinline__ void mat3mul(float* D, const float* X, const float* Y) {
    #pragma unroll
    for (int r = 0; r < 3; ++r)
        #pragma unroll
        for (int c = 0; c < 3; ++c)
            D[r*3+c] = X[r*3+0]*Y[0*3+c] + X[r*3+1]*Y[1*3+c] + X[r*3+2]*Y[2*3+c];
}

// ---------------------------------------------------------------------------
// Kernel 1: per-batch geometry.  diff_j = (K R K^-1) pix_j  (K t cancels),
// inv_dn_j = 1/||diff_j||, avec_i = src_i - K t.
// ---------------------------------------------------------------------------
__global__ void k_setup(const float* __restrict__ fsrc,
                        const float* __restrict__ Kp,
                        const float* __restrict__ Rp,
                        const float* __restrict__ tp,
                        float* __restrict__ diff,
                        float* __restrict__ inv_dn,
                        float* __restrict__ avec) {
    const int b = blockIdx.x;
    float K[9], R[9], t3[3];
    #pragma unroll
    for (int i = 0; i < 9; ++i) { K[i] = Kp[b*9+i]; R[i] = Rp[b*9+i]; }
    #pragma unroll
    for (int i = 0; i < 3; ++i) t3[i] = tp[b*3+i];

    // K^-1 via adjugate
    float c00 = K[4]*K[8] - K[5]*K[7];
    float c01 = K[5]*K[6] - K[3]*K[8];
    float c02 = K[3]*K[7] - K[4]*K[6];
    float id  = 1.0f / (K[0]*c00 + K[1]*c01 + K[2]*c02);
    float Ki[9];
    Ki[0] = c00*id;                    Ki[1] = (K[2]*K[7]-K[1]*K[8])*id;  Ki[2] = (K[1]*K[5]-K[2]*K[4])*id;
    Ki[3] = c01*id;                    Ki[4] = (K[0]*K[8]-K[2]*K[6])*id;  Ki[5] = (K[2]*K[3]-K[0]*K[5])*id;
    Ki[6] = c02*id;                    Ki[7] = (K[1]*K[6]-K[0]*K[7])*id;  Ki[8] = (K[0]*K[4]-K[1]*K[3])*id;

    float KR[9], M[9];
    mat3mul(KR, K, R);
    mat3mul(M, KR, Ki);                          // M = K R K^-1

    float o0 = K[0]*t3[0] + K[1]*t3[1] + K[2]*t3[2];   // o_proj = K t
    float o1 = K[3]*t3[0] + K[4]*t3[1] + K[5]*t3[2];
    float o2 = K[6]*t3[0] + K[7]*t3[1] + K[8]*t3[2];

    for (int n = threadIdx.x; n < NPIX; n += blockDim.x) {
        float h = (float)(n >> 6);               // meshgrid 'ij': pix = (h, w, 1)
        float w = (float)(n & 63);
        float d0 = M[0]*h + M[1]*w + M[2];
        float d1 = M[3]*h + M[4]*w + M[5];
        float d2 = M[6]*h + M[7]*w + M[8];
        diff[(b*3+0)*NPIX + n] = d0;
        diff[(b*3+1)*NPIX + n] = d1;
        diff[(b*3+2)*NPIX + n] = d2;
        inv_dn[b*NPIX + n] = fast_rsq(d0*d0 + d1*d1 + d2*d2);
        avec[(b*3+0)*NPIX + n] = fsrc[(b*3+0)*NPIX + n] - o0;
        avec[(b*3+1)*NPIX + n] = fsrc[(b*3+1)*NPIX + n] - o1;
        avec[(b*3+2)*NPIX + n] = fsrc[(b*3+2)*NPIX + n] - o2;
    }
}

// ---------------------------------------------------------------------------
// Kernel 2: per-row (i) online softmax stats of s[i,j] = 50*(d_epi - 0.5),
// j swept lane-strided, wave32 butterfly combine.
// ---------------------------------------------------------------------------
__global__ void k_rowstats(const float* __restrict__ diff,
                           const float* __restrict__ inv_dn,
                           const float* __restrict__ avec,
                           float* __restrict__ row_m,
                           float* __restrict__ row_l) {
    const int wave = (blockIdx.x * blockDim.x + threadIdx.x) >> 5;
    const int lane = threadIdx.x & 31;
    const int b = wave >> 12;          // 4096 rows per batch
    const int i = wave & 4095;

    const float a0 = avec[(b*3+0)*NPIX + i];
    const float a1 = avec[(b*3+1)*NPIX + i];
    const float a2 = avec[(b*3+2)*NPIX + i];

    float m = -3.0e38f, l = 0.0f;
    for (int j = lane; j < NPIX; j += 32) {
        float d0 = diff[(b*3+0)*NPIX + j];
        float d1 = diff[(b*3+1)*NPIX + j];
        float d2 = diff[(b*3+2)*NPIX + j];
        float c0 = a1*d2 - a2*d1;
        float c1 = a2*d0 - a0*d2;
        float c2 = a0*d1 - a1*d0;
        float dep = fast_sqrt(c0*c0 + c1*c1 + c2*c2) * inv_dn[b*NPIX + j];
        float s = 50.0f * (dep - 0.5f);
        float nm = fmaxf(m, s);
        l = l * __expf(m - nm) + __expf(s - nm);
        m = nm;
    }
    #pragma unroll
    for (int mask = 16; mask >= 1; mask >>= 1) {
        float om = __shfl_xor(m, mask, 32);
        float ol = __shfl_xor(l, mask, 32);
        softmax_combine(m, l, om, ol);
    }
    if (lane == 0) { row_m[b*NPIX + i] = m; row_l[b*NPIX + i] = l; }
}

// ---------------------------------------------------------------------------
// Kernel 3: per-column (k) online softmax stats of A_weighted[i,k].
// One wave per 16-column tile; loops over 16-row tiles.  A tile computed via
// WMMA with operands transposed so the D tile lands as D[m=k,n=i]:
//   Aop[m][c] = tar[c][k0+m]  (fixed per wave),  Bop[c][n] = src[c][i0+n].
// Lane l handles i = i0 + (l&15); vgpr r handles k = k0 + r + 8*(l>=16).
// ---------------------------------------------------------------------------
__global__ void k_colstats(const float* __restrict__ fsrc,
                           const float* __restrict__ ftar,
                           const float* __restrict__ diff,
                           const float* __restrict__ inv_dn,
                           const float* __restrict__ avec,
                           const float* __restrict__ row_m,
                           const float* __restrict__ row_l,
                           float* __restrict__ col_m,
                           float* __restrict__ col_l) {
    const int wave = (blockIdx.x * blockDim.x + threadIdx.x) >> 5;
    const int lane = threadIdx.x & 31;
    const int b  = wave >> 8;          // 256 k-tiles per batch
    const int k0 = (wave & 255) * 16;
    const int li = lane & 15;
    const int hi = lane >> 4;

    // A fragment (tar, 16 k-rows x K=4 padded)
    v2f afrag;
    afrag.x = ftar[(b*3 + (hi ? 2 : 0))*NPIX + k0 + li];
    afrag.y = hi ? 0.0f : ftar[(b*3+1)*NPIX + k0 + li];

    // per-lane geometry of its 8 owned columns k (fixed for whole wave)
    float kd0[8], kd1[8], kd2[8], kir[8], cm[8], cl[8];
    #pragma unroll
    for (int r = 0; r < 8; ++r) {
        int k = k0 + r + 8*hi;
        kd0[r] = diff[(b*3+0)*NPIX + k];
        kd1[r] = diff[(b*3+1)*NPIX + k];
        kd2[r] = diff[(b*3+2)*NPIX + k];
        kir[r] = inv_dn[b*NPIX + k];
        cm[r] = -3.0e38f; cl[r] = 0.0f;
    }

    for (int it = 0; it < 256; ++it) {
        const int i = it*16 + li;
        v2f bfrag;
        bfrag.x = fsrc[(b*3 + (hi ? 2 : 0))*NPIX + i];
        bfrag.y = hi ? 0.0f : fsrc[(b*3+1)*NPIX + i];
        v8f cz = {};
        v8f At = __builtin_amdgcn_wmma_f32_16x16x4_f32(
            false, afrag, false, bfrag, (short)0, cz, false, false);
        // row-i data (shared by lanes l and l+16)
        float a0 = avec[(b*3+0)*NPIX + i];
        float a1 = avec[(b*3+1)*NPIX + i];
        float a2 = avec[(b*3+2)*NPIX + i];
        float rm = row_m[b*NPIX + i];
        float irl = fast_rcp(row_l[b*NPIX + i]);
        #pragma unroll
        for (int r = 0; r < 8; ++r) {
            float c0 = a1*kd2[r] - a2*kd1[r];
            float c1 = a2*kd0[r] - a0*kd2[r];
            float c2 = a0*kd1[r] - a1*kd0[r];
            float dep = fast_sqrt(c0*c0 + c1*c1 + c2*c2) * kir[r];
            float s = 50.0f * (dep - 0.5f);
            float wgt = 1.0f - __expf(s - rm) * irl;    // 1 - softmax_j
            float Aw = At[r] * wgt;                     // A[i,k] * weight
            float nm = fmaxf(cm[r], Aw);
            cl[r] = cl[r] * __expf(cm[r] - nm) + __expf(Aw - nm);
            cm[r] = nm;
        }
    }
    // combine the 16 lanes of each half (each half owns disjoint k's)
    #pragma unroll
    for (int mask = 8; mask >= 1; mask >>= 1) {
        #pragma unroll
        for (int r = 0; r < 8; ++r) {
            float om = __shfl_xor(cm[r], mask, 32);
            float ol = __shfl_xor(cl[r], mask, 32);
            softmax_combine(cm[r], cl[r], om, ol);
        }
    }
    if (li == 0) {
        #pragma unroll
        for (int r = 0; r < 8; ++r) {
            int k = k0 + r + 8*hi;
            col_m[b*NPIX + k] = cm[r];
            col_l[b*NPIX + k] = cl[r];
        }
    }
}

// ---------------------------------------------------------------------------
// Kernel 4: output.  One wave per 16-row tile; loops k-tiles.  Standard
// orientation: Aop[m][c] = src[c][i0+m] (fixed), Bop[c][n] = tar[c][k0+n].
// attended[i,c] = sum_k softmax_i(Aw)[i,k] * src[c,k]; out[b,i*3+c] layout
// matches the reference's (N,C)->(C,H,W) flat reinterpretation.
// ---------------------------------------------------------------------------
__global__ void k_output(const float* __restrict__ fsrc,
                         const float* __restrict__ ftar,
                         const float* __restrict__ diff,
                         const float* __restrict__ inv_dn,
                         const float* __restrict__ avec,
                         const float* __restrict__ row_m,
                         const float* __restrict__ row_l,
                         const float* __restrict__ col_m,
                         const float* __restrict__ col_l,
                         float* __restrict__ out) {
    const int wave = (blockIdx.x * blockDim.x + threadIdx.x) >> 5;
    const int lane = threadIdx.x & 31;
    const int b  = wave >> 8;
    const int i0 = (wave & 255) * 16;
    const int li = lane & 15;
    const int hi = lane >> 4;

    // A fragment (src rows, 16 i-rows x K=4 padded), fixed per wave
    v2f afrag;
    afrag.x = fsrc[(b*3 + (hi ? 2 : 0))*NPIX + i0 + li];
    afrag.y = hi ? 0.0f : fsrc[(b*3+1)*NPIX + i0 + li];

    // per-lane row data of its 8 owned rows i (fixed for whole wave)
    float ra0[8], ra1[8], ra2[8], rm[8], irl[8];
    float acc0[8], acc1[8], acc2[8];
    #pragma unroll
    for (int r = 0; r < 8; ++r) {
        int i = i0 + r + 8*hi;
        ra0[r] = avec[(b*3+0)*NPIX + i];
        ra1[r] = avec[(b*3+1)*NPIX + i];
        ra2[r] = avec[(b*3+2)*NPIX + i];
        rm[r]  = row_m[b*NPIX + i];
        irl[r] = fast_rcp(row_l[b*NPIX + i]);
        acc0[r] = 0.0f; acc1[r] = 0.0f; acc2[r] = 0.0f;
    }

    for (int kt = 0; kt < 256; ++kt) {
        const int k = kt*16 + li;
        v2f bfrag;
        bfrag.x = ftar[(b*3 + (hi ? 2 : 0))*NPIX + k];
        bfrag.y = hi ? 0.0f : ftar[(b*3+1)*NPIX + k];
        v8f cz = {};
        v8f At = __builtin_amdgcn_wmma_f32_16x16x4_f32(
            false, afrag, false, bfrag, (short)0, cz, false, false);
        // column-k data (shared by lanes l and l+16)
        float d0 = diff[(b*3+0)*NPIX + k];
        float d1 = diff[(b*3+1)*NPIX + k];
        float d2 = diff[(b*3+2)*NPIX + k];
        float ir = inv_dn[b*NPIX + k];
        float cmk = col_m[b*NPIX + k];
        float icl = fast_rcp(col_l[b*NPIX + k]);
        float sv0 = fsrc[(b*3+0)*NPIX + k];
        float sv1 = fsrc[(b*3+1)*NPIX + k];
        float sv2 = fsrc[(b*3+2)*NPIX + k];
        #pragma unroll
        for (int r = 0; r < 8; ++r) {
            float c0 = ra1[r]*d2 - ra2[r]*d1;
            float c1 = ra2[r]*d0 - ra0[r]*d2;
            float c2 = ra0[r]*d1 - ra1[r]*d0;
            float dep = fast_sqrt(c0*c0 + c1*c1 + c2*c2) * ir;
            float s = 50.0f * (dep - 0.5f);
            float wgt = 1.0f - __expf(s - rm[r]) * irl[r];
            float p = __expf(At[r]*wgt - cmk) * icl;    // softmax over i (col-normalized)
            acc0[r] += p * sv0;
            acc1[r] += p * sv1;
            acc2[r] += p * sv2;
        }
    }
    // sum partial k-contributions across the 16 lanes of each half
    #pragma unroll
    for (int mask = 8; mask >= 1; mask >>= 1) {
        #pragma unroll
        for (int r = 0; r < 8; ++r) {
            acc0[r] += __shfl_xor(acc0[r], mask, 32);
            acc1[r] += __shfl_xor(acc1[r], mask, 32);
            acc2[r] += __shfl_xor(acc2[r], mask, 32);
        }
    }
    #pragma unroll
    for (int r = 0; r < 8; ++r) {
        if (li == r) {
            int i = i0 + r + 8*hi;
            out[b*(NPIX*3) + i*3 + 0] = acc0[r];
            out[b*(NPIX*3) + i*3 + 1] = acc1[r];
            out[b*(NPIX*3) + i*3 + 2] = acc2[r];
        }
    }
}

// ---------------------------------------------------------------------------
extern "C" void kernel_launch(void* const* d_in, const int* in_sizes, int n_in,
                              void* d_out, int out_size, void* d_ws, size_t ws_size,
                              hipStream_t stream) {
    (void)in_sizes; (void)n_in; (void)out_size; (void)ws_size;
    const float* f_tar = (const float*)d_in[0];
    const float* f_src = (const float*)d_in[1];
    const float* Kp    = (const float*)d_in[2];
    const float* Rp    = (const float*)d_in[3];
    const float* tp    = (const float*)d_in[4];
    float* out = (float*)d_out;

    float* ws     = (float*)d_ws;          // all f32 scratch, ~352 KB total
    float* diff   = ws;                    // B*3*N = 24576
    float* inv_dn = ws + 24576;            // B*N   =  8192
    float* avec   = ws + 32768;            // B*3*N = 24576
    float* row_m  = ws + 57344;            // B*N
    float* row_l  = ws + 65536;            // B*N
    float* col_m  = ws + 73728;            // B*N
    float* col_l  = ws + 81920;            // B*N

    k_setup   <<<BATCH, 256, 0, stream>>>(f_src, Kp, Rp, tp, diff, inv_dn, avec);
    k_rowstats<<<(BATCH*NPIX*32)/256, 256, 0, stream>>>(diff, inv_dn, avec, row_m, row_l);
    k_colstats<<<(BATCH*256*32)/256, 256, 0, stream>>>(f_src, f_tar, diff, inv_dn, avec,
                                                       row_m, row_l, col_m, col_l);
    k_output  <<<(BATCH*256*32)/256, 256, 0, stream>>>(f_src, f_tar, diff, inv_dn, avec,
                                                       row_m, row_l, col_m, col_l, out);
}